// DynamicSegmentationGating_76287209111797
// MI455X (gfx1250) — compile-verified
//
#include <hip/hip_runtime.h>
#include <hip/hip_bf16.h>

typedef float v2f __attribute__((ext_vector_type(2)));
typedef float v4f __attribute__((ext_vector_type(4)));
typedef float v8f __attribute__((ext_vector_type(8)));
typedef unsigned int u32x4 __attribute__((ext_vector_type(4)));
typedef int i32x4 __attribute__((ext_vector_type(4)));
typedef int i32x8 __attribute__((ext_vector_type(8)));

#define BATCH 64
#define SEQ   4096
#define DIM   128
#define TOK_PER_BLOCK 64          // 4 waves x 16 tokens
#define LDS_STRIDE 132            // 128 + 4 pad -> conflict-free ds_load_b64 in WMMA A layout

// ---------------------------------------------------------------------------
// Kernel 1: per-token dot product  s[t] = z[t,:] . W   via V_WMMA_F32_16X16X4_F32,
// tile staged global->LDS by the Tensor Data Mover (tensor_load_to_lds).
// ---------------------------------------------------------------------------
__global__ __launch_bounds__(128) void dsg_score_kernel(const float* __restrict__ z,
                                                        const float* __restrict__ W,
                                                        float* __restrict__ s)
{
    __shared__ float ldsZ[TOK_PER_BLOCK * LDS_STRIDE];
    __shared__ float ldsW[DIM];

    const int tid = threadIdx.x;
    const size_t tokBase = (size_t)blockIdx.x * TOK_PER_BLOCK;   // global token index in [0, B*L)
    const float* gz = z + tokBase * DIM;

    if (tid < DIM) ldsW[tid] = W[tid];

#if defined(__gfx1250__) && __has_builtin(__builtin_amdgcn_tensor_load_to_lds)
    // ---- TDM path: one DMA of the 64x128 f32 tile, HW-padded to stride 132 ----
    if (tid < 32) {
        const unsigned lds_addr = (unsigned)(unsigned long long)(uintptr_t)&ldsZ[0]; // flat[31:0] = LDS offset
        const unsigned long long ga = (unsigned long long)(uintptr_t)gz;

        // D# group 0: count=1 | lds_addr | global_addr[56:0] | type=2
        u32x4 g0;
        g0[0] = 1u;                                   // count=1, user descriptor
        g0[1] = lds_addr;                             // LDS byte address
        g0[2] = (unsigned)(ga & 0xFFFFFFFFull);       // global_addr[31:0]
        g0[3] = (unsigned)((ga >> 32) & 0x01FFFFFFull) | (2u << 30); // [120:96]=addr[56:32], type=2

        // D# group 1: data_size=4B, pad every 128 DWORDs by 4 DWORDs,
        // tensor 128 x 64, tile 128 x 64, dim0 stride 128 (data_size units)
        i32x8 g1;
        g1[0] = (int)((2u << 16)          // data_size = 2 -> 4 bytes
                    | (1u << 20)          // pad_enable
                    | (6u << 22)          // pad_interval = 6 -> 128 DWORDs
                    | (3u << 25));        // pad_amount  = 3 -> 4 DWORDs
        g1[1] = (int)(128u << 16);        // tensor_dim0[15:0] = 128 (bits 63:48)
        g1[2] = (int)(64u << 16);         // tensor_dim0[31:16]=0, tensor_dim1[15:0]=64 (bits 95:80)
        g1[3] = (int)(128u << 16);        // tensor_dim1[31:16]=0, tile_dim0 = 128 (bits 127:112)
        g1[4] = 64;                       // tile_dim1 = 64, tile_dim2 = 0
        g1[5] = 128;                      // tensor_dim0_stride[31:0] = 128
        g1[6] = 0;                        // stride hi + tensor_dim1_stride lo
        g1[7] = 0;                        // tensor_dim1_stride hi

        i32x4 g2 = {0, 0, 0, 0};          // unused (2D tile)
        i32x4 g3 = {0, 0, 0, 0};

#if defined(__clang_major__) && __clang_major__ >= 23
        i32x8 g4 = {0, 0, 0, 0, 0, 0, 0, 0};
        __builtin_amdgcn_tensor_load_to_lds(g0, g1, g2, g3, g4, 0);
#else
        __builtin_amdgcn_tensor_load_to_lds(g0, g1, g2, g3, 0);
#endif
#if __has_builtin(__builtin_amdgcn_s_wait_tensorcnt)
        __builtin_amdgcn_s_wait_tensorcnt(0);
#else
        asm volatile("s_wait_tensorcnt 0x0" ::: "memory");
#endif
    }
    __syncthreads();
#else
    // ---- Fallback: coalesced cooperative load (also used for host parse) ----
#pragma unroll
    for (int i = 0; i < 16; ++i) {
        int idx = tid + i * 128;          // 0..2047
        int row = idx >> 5;               // token within tile
        int c4  = idx & 31;               // float4 column
        v4f v = *(const v4f*)(gz + (size_t)row * DIM + c4 * 4);
        *(v4f*)(&ldsZ[row * LDS_STRIDE + c4 * 4]) = v;
    }
    __syncthreads();
#endif

    const int wave = tid >> 5;            // 0..3 -> owns tokens [wave*16, wave*16+16)
    const int lane = tid & 31;
    const int half = lane >> 4;           // 0: K={0,1}, 1: K={2,3} per f32 A-matrix layout
    const float* zrow = &ldsZ[(wave * 16 + (lane & 15)) * LDS_STRIDE];
    float* sOut = s + tokBase;

#if defined(__gfx1250__) && __has_builtin(__builtin_amdgcn_wmma_f32_16x16x4_f32)
    v8f acc = {};
#pragma unroll
    for (int c = 0; c < 32; ++c) {
        const int k0 = c * 4 + half * 2;
        // A (16x4 f32): lanes 0-15 hold M=lane K=0/1; lanes 16-31 hold M=lane-16 K=2/3
        v2f a;
        a.x = zrow[k0];
        a.y = zrow[k0 + 1];
        // B (4x16 f32): broadcast W chunk across all 16 columns (every column identical)
        v2f bm;
        bm.x = ldsW[k0];
        bm.y = ldsW[k0 + 1];
        acc = __builtin_amdgcn_wmma_f32_16x16x4_f32(
            /*neg_a=*/false, a, /*neg_b=*/false, bm,
            /*c_mod=*/(short)0, acc, /*reuse_a=*/false, /*reuse_b=*/false);
    }
    // C/D layout: VGPR r holds M=r (lanes 0-15) and M=r+8 (lanes 16-31).
    // All columns equal -> lane 0 has s[tok 0..7], lane 16 has s[tok 8..15].
    if (lane == 0 || lane == 16) {
        const int t0 = wave * 16 + half * 8;
#pragma unroll
        for (int r = 0; r < 8; ++r) sOut[t0 + r] = acc[r];
    }
#else
    // VALU fallback (also used for the host-side parse): each lane pair splits the 128-dim dot
    float sum = 0.0f;
    const int kbeg = half * 64;
#pragma unroll 8
    for (int k = kbeg; k < kbeg + 64; ++k) sum += zrow[k] * ldsW[k];
    sum += __shfl_xor(sum, 16, 32);
    if (lane < 16) sOut[wave * 16 + lane] = sum;
#endif
}

// ---------------------------------------------------------------------------
// Kernel 2: per-batch-row sigmoid + boundary scan -> (probs, seg_ids)
// boundary[l] = sigmoid(s[l+1]-s[l]+b) > 0.5  <=>  (s[l+1]-s[l]+b) > 0
// seg[0] = 0 ; seg[j] = inclusive_cumsum(boundary)[j-1]
// ---------------------------------------------------------------------------
__global__ __launch_bounds__(256) void dsg_scan_kernel(const float* __restrict__ s,
                                                       const float* __restrict__ bias_p,
                                                       float* __restrict__ probs,
                                                       int* __restrict__ seg)
{
    __shared__ int lds[256];
    __shared__ int carryS;

    const int b = blockIdx.x;
    const int t = threadIdx.x;
    const float bias = bias_p[0];
    const float* srow = s + (size_t)b * SEQ;
    float* prow = probs + (size_t)b * (SEQ - 1);
    int* segrow = seg + (size_t)b * SEQ;

    if (t == 0) { carryS = 0; segrow[0] = 0; }
    __syncthreads();

    for (int chunk = 0; chunk < SEQ / 256; ++chunk) {
        const int l = chunk * 256 + t;    // boundary index, valid when l < SEQ-1
        int bnd = 0;
        if (l < SEQ - 1) {
            float x = srow[l + 1] - srow[l] + bias;
            float p = 1.0f / (1.0f + expf(-x));
            prow[l] = p;
            bnd = (x > 0.0f) ? 1 : 0;
        }
        // Hillis-Steele inclusive scan over 256 threads
        int v = bnd;
        lds[t] = v;
        __syncthreads();
#pragma unroll
        for (int off = 1; off < 256; off <<= 1) {
            int add = (t >= off) ? lds[t - off] : 0;
            __syncthreads();
            v += add;
            lds[t] = v;
            __syncthreads();
        }
        const int incl = v + carryS;      // inclusive cumsum of boundary[0..l]
        if (l < SEQ - 1) segrow[l + 1] = incl;
        __syncthreads();
        if (t == 255) carryS = incl;
        __syncthreads();
    }
}

// ---------------------------------------------------------------------------
extern "C" void kernel_launch(void* const* d_in, const int* in_sizes, int n_in,
                              void* d_out, int out_size, void* d_ws, size_t ws_size,
                              hipStream_t stream)
{
    const float* z    = (const float*)d_in[0];   // (64, 4096, 128) f32
    const float* W    = (const float*)d_in[1];   // (128, 1) f32
    const float* bias = (const float*)d_in[2];   // (1,) f32

    float* scores = (float*)d_ws;                       // 64*4096 f32 = 1 MB scratch
    float* probs  = (float*)d_out;                      // (64, 4095) f32
    int*   seg    = (int*)(probs + (size_t)BATCH * (SEQ - 1));  // (64, 4096) i32 (bitwise)

    const int n_tokens = BATCH * SEQ;
    dsg_score_kernel<<<n_tokens / TOK_PER_BLOCK, 128, 0, stream>>>(z, W, scores);
    dsg_scan_kernel<<<BATCH, 256, 0, stream>>>(scores, bias, probs, seg);
}